// PWLUBase_23742579212531
// MI455X (gfx1250) — compile-verified
//
#include <hip/hip_runtime.h>
#include <hip/hip_bf16.h>

// PWLU forward, channelwise. x: [B, C, H, W] fp32, per-channel params.
// Memory-bound streaming kernel: ~205 MB traffic -> ~8.8 us floor @ 23.3 TB/s.
//
// Per channel c (n_regions = 6, n_points = 7):
//   rl       = (rb - lb) / 6
//   sim_left = lb - rl
//   xn       = (x - sim_left) / (7 * rl)
//   r        = floor(clip(xn, 0, 1.001) * 7)           // 0..7
//   dist     = (xn*7 - r) * rl
//   out      = fp[r] + dist * s[r]
// with fp/s the 8-entry false_points/slopes tables from the reference.

#define N_POINTS 7
#define HW 3136      // 56*56 (fixed by reference)
#define HW4 784      // HW / 4 = 3*256 + 16

typedef float v4f __attribute__((ext_vector_type(4)));
typedef float v2f __attribute__((ext_vector_type(2)));

__global__ __launch_bounds__(256) void pwlu_fwd_kernel(
    const float* __restrict__ x,
    const float* __restrict__ points,       // [C, 7]
    const float* __restrict__ bounds,       // [C, 2]
    const float* __restrict__ left_slopes,  // [C]
    const float* __restrict__ right_slopes, // [C]
    float* __restrict__ out,
    int C, int cmask)                       // cmask = C-1 if C is pow2, else -1
{
    __shared__ v2f   tab[8];    // (left_point, slope) per region index
    __shared__ float chp[2];    // [0] = sim_left, [1] = region_len

    const int plane = blockIdx.x;                       // plane = b*C + c
    const int c     = (cmask >= 0) ? (plane & cmask) : (plane % C);

    if (threadIdx.x == 0) {
        const float lb = bounds[2 * c + 0];
        const float rb = bounds[2 * c + 1];
        const float rl = (rb - lb) * (1.0f / 6.0f);
        const float inv_rl = 1.0f / rl;
        const float ls = left_slopes[c];
        const float rs = right_slopes[c];

        float prev = points[(size_t)c * N_POINTS + 0];
        v2f t0; t0.x = fmaf(-ls, rl, prev); t0.y = ls;   // fp0 = p0 - ls*rl
        tab[0] = t0;
#pragma unroll
        for (int i = 1; i < N_POINTS; ++i) {
            const float pi = points[(size_t)c * N_POINTS + i];
            v2f ti; ti.x = prev; ti.y = (pi - prev) * inv_rl;
            tab[i] = ti;
            prev = pi;
        }
        v2f t7; t7.x = prev; t7.y = rs;                  // last point, right slope
        tab[7] = t7;
        chp[0] = lb - rl;                                // sim_left
        chp[1] = rl;
    }
    __syncthreads();

    const float sim_left = chp[0];
    const float rl       = chp[1];
    const float scale    = 1.0f / (7.0f * rl);           // 1 / ((n_regions+1)*rl)

    const v4f* __restrict__ xin = (const v4f*)(x   + (size_t)plane * HW);
    v4f*       __restrict__ yo  = (v4f*)      (out + (size_t)plane * HW);

    const int tid = threadIdx.x;

    // 3 full strides + 16-thread tail (784 = 3*256 + 16). Issue all three b128
    // NT loads up front for per-wave MLP, then compute, then store.
    v4f v0 = __builtin_nontemporal_load(&xin[tid        ]);
    v4f v1 = __builtin_nontemporal_load(&xin[tid + 256  ]);
    v4f v2 = __builtin_nontemporal_load(&xin[tid + 512  ]);

    auto eval4 = [&](v4f v) -> v4f {
        v4f r;
#pragma unroll
        for (int k = 0; k < 4; ++k) {
            const float xn  = (v[k] - sim_left) * scale;
            const float cl  = fminf(fmaxf(xn, 0.0f), 1.001f);
            const float reg = floorf(cl * 7.0f);
            const int   ri  = (int)reg;                   // 0..7
            const float d   = fmaf(xn, 7.0f, -reg) * rl;
            const v2f   t   = tab[ri];                    // ds_load_b64
            r[k] = fmaf(d, t.y, t.x);
        }
        return r;
    };

    v4f r0 = eval4(v0);
    v4f r1 = eval4(v1);
    v4f r2 = eval4(v2);
    __builtin_nontemporal_store(r0, &yo[tid      ]);
    __builtin_nontemporal_store(r1, &yo[tid + 256]);
    __builtin_nontemporal_store(r2, &yo[tid + 512]);

    if (tid < HW4 - 768) {                                // 16-thread tail
        v4f v3 = __builtin_nontemporal_load(&xin[tid + 768]);
        v4f r3 = eval4(v3);
        __builtin_nontemporal_store(r3, &yo[tid + 768]);
    }
}

extern "C" void kernel_launch(void* const* d_in, const int* in_sizes, int n_in,
                              void* d_out, int out_size, void* d_ws, size_t ws_size,
                              hipStream_t stream) {
    const float* x      = (const float*)d_in[0];
    const float* points = (const float*)d_in[1];
    const float* bounds = (const float*)d_in[2];
    const float* ls     = (const float*)d_in[3];
    const float* rs     = (const float*)d_in[4];
    float* out          = (float*)d_out;

    const int C      = in_sizes[3];                    // left_slopes has C elements
    const int planes = in_sizes[0] / HW;               // B * C
    const int cmask  = ((C & (C - 1)) == 0) ? (C - 1) : -1;

    pwlu_fwd_kernel<<<planes, 256, 0, stream>>>(x, points, bounds, ls, rs, out, C, cmask);
}